// Bessel_50809463112381
// MI455X (gfx1250) — compile-verified
//
#include <hip/hip_runtime.h>

typedef __attribute__((ext_vector_type(2))) float v2f;
typedef __attribute__((ext_vector_type(8))) float v8f;

// Problem constants (fixed by the reference): x:(8192,64) y:(8192,64), out:(8192,8192)
constexpr int NROW = 8192;   // rows of x  (M)
constexpr int NCOL = 8192;   // rows of y  (N)
constexpr int DDIM = 64;     // K
constexpr float AA = 10.0f;  // module arg a
constexpr float L2E = 1.44269504088896340736f;  // log2(e)

// ---------------------------------------------------------------------------
// Cold path: full-precision i0e(z)*exp(z-2a) for any dot in [-1,1].
// Only reached if some lane in the wave has dot <= -0.6 (~5 sigma event).
// ---------------------------------------------------------------------------
__device__ __noinline__ float bessel_slow(float raw) {
    float dot = __builtin_amdgcn_fmed3f(raw, -1.0f, 1.0f);
    float u   = fmaxf(__builtin_fmaf(0.5f, dot, 0.5f), 1e-12f);
    float z   = (2.0f * AA) * __fsqrt_rn(u);
    float i0e;
    if (z <= 8.0f) {
        const float As[18] = {
            -1.30002500998624804212e-8f,  6.04699502254191894932e-8f,
            -2.67079385394061173391e-7f,  1.11738753912010371815e-6f,
            -4.41673835845875056359e-6f,  1.64484480707288970893e-5f,
            -5.75419501008210370398e-5f,  1.88502885095841655729e-4f,
            -5.76375574538582365885e-4f,  1.63947561694133579842e-3f,
            -4.32430999505057594430e-3f,  1.05464603945949983183e-2f,
            -2.37374148058994688156e-2f,  4.93052842396707084878e-2f,
            -9.49010970480476444210e-2f,  1.71620901522208775349e-1f,
            -3.04682672343198398683e-1f,  6.76795274409476084995e-1f };
        float ys = __builtin_fmaf(0.5f, z, -2.0f);
        float s0 = 0.0f, s1 = 0.0f, s2 = 0.0f;
        #pragma unroll
        for (int i = 0; i < 18; ++i) {
            s2 = s1; s1 = s0;
            s0 = __builtin_fmaf(ys, s1, As[i] - s2);
        }
        i0e = 0.5f * (s0 - s2);
    } else {
        const float Bs[8] = {
             4.94060238822496958910e-10f, 3.39623202570838634515e-9f,
             2.26666899049817806459e-8f,  2.04891858946906374183e-7f,
             2.89137052083475648297e-6f,  6.88975834691682398426e-5f,
             3.36911647825569408990e-3f,  8.04490411014108831608e-1f };
        float yl = __builtin_fmaf(32.0f, __frcp_rn(z), -2.0f);
        float l0 = 0.0f, l1 = 0.0f, l2 = 0.0f;
        #pragma unroll
        for (int i = 0; i < 8; ++i) {
            l2 = l1; l1 = l0;
            l0 = __builtin_fmaf(yl, l1, Bs[i] - l2);
        }
        i0e = 0.5f * (l0 - l2) * __frsqrt_rn(z);
    }
    return i0e * __expf(z - 2.0f * AA);
}

// ---------------------------------------------------------------------------
// Fast path, PACKED over two elements (targets v_pk_fma_f32 / v_pk_mul_f32).
// Valid for u >= 0.2 (z >= 8.94, guaranteed by the wave-uniform guard).
//   r = rsqrt(u); z = (20u)*r; 32/z-2 = 1.6r-2; rsqrt(z) = sqrt(r)/sqrt(20)
//   0.5/sqrt(20) folded into Chebyshev coeffs; exp(z-20)=exp2(z*L2E - 20*L2E).
// ---------------------------------------------------------------------------
__device__ __forceinline__ v2f bessel_fast2(float ra, float rb) {
    constexpr float SCL = 0.11180339887498949f;  // 0.5 / sqrt(20)
    const float Bs[6] = {
        2.26666899049817806459e-8f * SCL,  2.04891858946906374183e-7f * SCL,
        2.89137052083475648297e-6f * SCL,  6.88975834691682398426e-5f * SCL,
        3.36911647825569408990e-3f * SCL,  8.04490411014108831608e-1f * SCL };

    v2f d;
    d.x = __builtin_amdgcn_fmed3f(ra, -1.0f, 1.0f);    // scalar: no pk med3
    d.y = __builtin_amdgcn_fmed3f(rb, -1.0f, 1.0f);
    v2f u = d * 0.5f + 0.5f;                           // pk_fma
    v2f r;
    r.x = __builtin_amdgcn_rsqf(u.x);                  // TRANS
    r.y = __builtin_amdgcn_rsqf(u.y);                  // TRANS
    v2f z  = (u * (2.0f * AA)) * r;                    // 2x pk_mul: 20*sqrt(u)
    v2f yl = r * (32.0f / (2.0f * AA)) - 2.0f;         // pk_fma: 32/z - 2
    v2f l0 = {0.0f, 0.0f}, l1 = {0.0f, 0.0f}, l2 = {0.0f, 0.0f};
    #pragma unroll
    for (int i = 0; i < 6; ++i) {
        l2 = l1; l1 = l0;
        l0 = yl * l1 + (Bs[i] - l2);                   // pk_add + pk_fma
    }
    v2f s;
    s.x = __builtin_amdgcn_sqrtf(r.x);                 // TRANS: u^(-1/4)
    s.y = __builtin_amdgcn_sqrtf(r.y);                 // TRANS
    v2f w = z * L2E - (2.0f * AA * L2E);               // pk_fma
    v2f e;
    e.x = __builtin_amdgcn_exp2f(w.x);                 // TRANS
    e.y = __builtin_amdgcn_exp2f(w.y);                 // TRANS
    return ((l0 - l2) * s) * e;                        // pk_add + 2x pk_mul
}

// Each wave: one 16-row M tile x NT=4 16-col N tiles.
constexpr int NT = 4;

__global__ __launch_bounds__(256)
void Bessel_50809463112381_kernel(const float* __restrict__ x,
                                  const float* __restrict__ y,
                                  float* __restrict__ out) {
    const int lane   = threadIdx.x & 31;
    const int waveid = (blockIdx.x << 3) | (threadIdx.x >> 5);

    // 512 M-tiles, 128 N-groups of 64 columns
    const int mt = waveid >> 7;        // 0..511
    const int ng = waveid & 127;       // 0..127
    const int m0 = mt * 16;
    const int n0 = ng * (16 * NT);

    const int lm = lane & 15;          // row/col within tile
    const int hk = (lane >> 4) * 2;    // K sub-offset 0 or 2 (half-wave split)

    // ---- preload A tile: x[m0+lm][0..63], float2 per k-step ----
    const float* xrow = x + (size_t)(m0 + lm) * DDIM + hk;
    v2f a[16];
    #pragma unroll
    for (int k = 0; k < 16; ++k)
        a[k] = *(const v2f*)(xrow + k * 4);

    v8f acc[NT];
    #pragma unroll
    for (int t = 0; t < NT; ++t) {
        const float* yrow = y + (size_t)(n0 + t * 16 + lm) * DDIM + hk;
        v8f c = {};
        #pragma unroll
        for (int k = 0; k < 16; ++k) {
            v2f b = *(const v2f*)(yrow + k * 4);
            // D = A(16x4 f32) * B(4x16 f32) + C(16x16 f32)
            c = __builtin_amdgcn_wmma_f32_16x16x4_f32(
                    /*neg_a=*/false, a[k],
                    /*neg_b=*/false, b,
                    /*c_mod=*/(short)0, c,
                    /*reuse_a=*/false, /*reuse_b=*/false);
        }
        acc[t] = c;
    }

    // ---- one wave-uniform rarity check for the whole 16x64 strip ----
    float dmin = acc[0][0];
    #pragma unroll
    for (int t = 0; t < NT; ++t)
        #pragma unroll
        for (int j = 0; j < 8; ++j)
            dmin = fminf(dmin, acc[t][j]);
    const bool rare = __builtin_amdgcn_ballot_w32(dmin <= -0.6f) != 0u;

    // ---- epilogue: z = 2a*sqrt((1+dot)/2); out = i0e(z)*exp(z-2a) ----
    const int rbase = m0 + ((lane >> 4) << 3);   // M = j + 8*(lane>=16)
    if (!rare) {
        #pragma unroll
        for (int t = 0; t < NT; ++t) {
            const int col = n0 + t * 16 + lm;
            #pragma unroll
            for (int jj = 0; jj < 4; ++jj) {     // row pairs (2jj, 2jj+1)
                v2f res = bessel_fast2(acc[t][2 * jj], acc[t][2 * jj + 1]);
                const size_t i0 = (size_t)(rbase + 2 * jj) * NCOL + col;
                __builtin_nontemporal_store(res.x, out + i0);
                __builtin_nontemporal_store(res.y, out + i0 + NCOL);
            }
        }
    } else {
        #pragma unroll
        for (int t = 0; t < NT; ++t) {
            const int col = n0 + t * 16 + lm;
            #pragma unroll
            for (int j = 0; j < 8; ++j) {
                const size_t idx = (size_t)(rbase + j) * NCOL + col;
                __builtin_nontemporal_store(bessel_slow(acc[t][j]), out + idx);
            }
        }
    }
}

extern "C" void kernel_launch(void* const* d_in, const int* in_sizes, int n_in,
                              void* d_out, int out_size, void* d_ws, size_t ws_size,
                              hipStream_t stream) {
    const float* x = (const float*)d_in[0];
    const float* y = (const float*)d_in[1];
    float* out = (float*)d_out;

    // waves = (8192/16) * (8192/64) = 512*128 = 65536; 8 waves/block -> 8192 blocks
    dim3 grid(8192), block(256);
    Bessel_50809463112381_kernel<<<grid, block, 0, stream>>>(x, y, out);
}